// OntologyEmbedding_27805618275280
// MI455X (gfx1250) — compile-verified
//
#include <hip/hip_runtime.h>
#include <math.h>

#define N_NODES 100000
#define N_EDGES 1600000
#define N_VOCAB 50000
#define DIM 128
#define NEG_SLOPE 0.2f

typedef __attribute__((ext_vector_type(2))) float v2f;
typedef __attribute__((ext_vector_type(8))) float v8f;

// ---------------------------------------------------------------------------
// xw = x @ W   (x: [N,128] row-major, W: [128,128] row-major)
// One block = 16-row strip (staged in LDS), one wave32 per 16-col tile.
// Inner loop: V_WMMA_F32_16X16X4_F32, K stepped by 4 (32 WMMAs / tile).
// ---------------------------------------------------------------------------
__global__ __launch_bounds__(256) void gemm_xw_kernel(const float* __restrict__ x,
                                                      const float* __restrict__ W,
                                                      float* __restrict__ xw) {
    __shared__ float ldsA[16 * DIM];  // 8 KB
    const int rowBase = blockIdx.x * 16;
    const int t = threadIdx.x;

    // cooperative stage of the 16x128 A strip (2 x float4 per thread)
    {
        const float4* src = (const float4*)(x + (size_t)rowBase * DIM);
        float4* dst4 = (float4*)ldsA;
        dst4[t]       = src[t];
        dst4[t + 256] = src[t + 256];
    }
    __syncthreads();

    const int lane    = t & 31;
    const int colBase = (t >> 5) * 16;       // wave id -> column tile
    const int mrow    = lane & 15;           // A: M = lane%16
    const int khalf   = (lane >> 4) * 2;     // A/B: lanes 16-31 hold K+2,K+3
    const int ncol    = colBase + (lane & 15);

    v8f c = {};
    for (int k = 0; k < DIM; k += 4) {
        v2f a, b;
        a.x = ldsA[mrow * DIM + k + khalf];
        a.y = ldsA[mrow * DIM + k + khalf + 1];
        b.x = W[(k + khalf) * DIM + ncol];
        b.y = W[(k + khalf + 1) * DIM + ncol];
        c = __builtin_amdgcn_wmma_f32_16x16x4_f32(false, a, false, b,
                                                  (short)0, c, false, false);
    }

    // D layout: VGPR r -> M=r (lanes 0-15), M=r+8 (lanes 16-31); N = lane%16
    const int rowOff = (lane >> 4) * 8;
#pragma unroll
    for (int r = 0; r < 8; ++r) {
        xw[(size_t)(rowBase + rowOff + r) * DIM + ncol] = c[r];
    }
}

// ---------------------------------------------------------------------------
// Per-node attention scalars: s_src = xw . a_src, s_dst = xw . a_dst
// One wave32 per node, float4 per lane, wave shuffle reduction.
// ---------------------------------------------------------------------------
__global__ __launch_bounds__(256) void proj_kernel(const float* __restrict__ xw,
                                                   const float* __restrict__ a_src,
                                                   const float* __restrict__ a_dst,
                                                   float* __restrict__ s_src,
                                                   float* __restrict__ s_dst) {
    const int node = (blockIdx.x * blockDim.x + threadIdx.x) >> 5;
    const int lane = threadIdx.x & 31;
    if (node >= N_NODES) return;
    const float4 xv = ((const float4*)(xw + (size_t)node * DIM))[lane];
    const float4 as = ((const float4*)a_src)[lane];
    const float4 ad = ((const float4*)a_dst)[lane];
    float vs = xv.x * as.x + xv.y * as.y + xv.z * as.z + xv.w * as.w;
    float vd = xv.x * ad.x + xv.y * ad.y + xv.z * ad.z + xv.w * ad.w;
#pragma unroll
    for (int off = 16; off > 0; off >>= 1) {
        vs += __shfl_xor(vs, off, 32);
        vd += __shfl_xor(vd, off, 32);
    }
    if (lane == 0) { s_src[node] = vs; s_dst[node] = vd; }
}

// ---------------------------------------------------------------------------
// init: segment-max encoding to ordered(-inf), segment-sum to 0,
// output accumulator pre-loaded with bias (folds the "+ bias" epilogue).
// ---------------------------------------------------------------------------
__global__ void init_kernel(float* __restrict__ h, const float* __restrict__ bias,
                            unsigned* __restrict__ m_ord, float* __restrict__ ssum) {
    const int i = blockIdx.x * blockDim.x + threadIdx.x;
    if (i < N_NODES) { m_ord[i] = 0x007FFFFFu; ssum[i] = 0.0f; }  // enc(-inf)
    if (i < N_NODES * DIM) h[i] = bias[i & (DIM - 1)];
}

// order-preserving float <-> uint mapping so segment-max can use atomicMax(u32)
__device__ __forceinline__ unsigned ord_enc(float f) {
    unsigned u = __float_as_uint(f);
    return (u & 0x80000000u) ? ~u : (u | 0x80000000u);
}
__device__ __forceinline__ float ord_dec(unsigned u) {
    return __uint_as_float((u & 0x80000000u) ? (u ^ 0x80000000u) : ~u);
}
__device__ __forceinline__ float lrelu(float x) { return x > 0.0f ? x : NEG_SLOPE * x; }

// ---------------------------------------------------------------------------
// edge pass 1: segment max of attention logits (self-loops appended)
// ---------------------------------------------------------------------------
__global__ void edge_max_kernel(const int* __restrict__ esrc, const int* __restrict__ edst,
                                const float* __restrict__ s_src, const float* __restrict__ s_dst,
                                unsigned* __restrict__ m_ord) {
    const int e = blockIdx.x * blockDim.x + threadIdx.x;
    if (e >= N_EDGES + N_NODES) return;
    const int s = (e < N_EDGES) ? esrc[e] : (e - N_EDGES);
    const int d = (e < N_EDGES) ? edst[e] : (e - N_EDGES);
    const float logit = lrelu(s_src[s] + s_dst[d]);
    atomicMax(&m_ord[d], ord_enc(logit));
}

// ---------------------------------------------------------------------------
// edge pass 2: segment sum of exp(logit - max)
// ---------------------------------------------------------------------------
__global__ void edge_sum_kernel(const int* __restrict__ esrc, const int* __restrict__ edst,
                                const float* __restrict__ s_src, const float* __restrict__ s_dst,
                                const unsigned* __restrict__ m_ord, float* __restrict__ ssum) {
    const int e = blockIdx.x * blockDim.x + threadIdx.x;
    if (e >= N_EDGES + N_NODES) return;
    const int s = (e < N_EDGES) ? esrc[e] : (e - N_EDGES);
    const int d = (e < N_EDGES) ? edst[e] : (e - N_EDGES);
    const float logit = lrelu(s_src[s] + s_dst[d]);
    atomicAdd(&ssum[d], __expf(logit - ord_dec(m_ord[d])));
}

// ---------------------------------------------------------------------------
// edge pass 3: h[dst] += alpha * xw[src]   (one wave32 per edge, float4/lane,
// 4 native global_atomic_add_f32 per lane)
// ---------------------------------------------------------------------------
__global__ __launch_bounds__(256) void edge_msg_kernel(
    const int* __restrict__ esrc, const int* __restrict__ edst,
    const float* __restrict__ s_src, const float* __restrict__ s_dst,
    const unsigned* __restrict__ m_ord, const float* __restrict__ ssum,
    const float* __restrict__ xw, float* __restrict__ h) {
    const long tid = (long)blockIdx.x * blockDim.x + threadIdx.x;
    const int e = (int)(tid >> 5);
    const int lane = threadIdx.x & 31;
    if (e >= N_EDGES + N_NODES) return;
    const int s = (e < N_EDGES) ? esrc[e] : (e - N_EDGES);
    const int d = (e < N_EDGES) ? edst[e] : (e - N_EDGES);
    const float logit = lrelu(s_src[s] + s_dst[d]);
    const float alpha = __expf(logit - ord_dec(m_ord[d])) / ssum[d];
    const float4 xv = ((const float4*)(xw + (size_t)s * DIM))[lane];
    float* out = h + (size_t)d * DIM + lane * 4;
    atomicAdd(out + 0, alpha * xv.x);
    atomicAdd(out + 1, alpha * xv.y);
    atomicAdd(out + 2, alpha * xv.z);
    atomicAdd(out + 3, alpha * xv.w);
}

// ---------------------------------------------------------------------------
// final vocabulary gather: out[v] = h[idx[v]]
// ---------------------------------------------------------------------------
__global__ __launch_bounds__(256) void gather_kernel(const int* __restrict__ idx,
                                                     const float* __restrict__ h,
                                                     float* __restrict__ out) {
    const long tid = (long)blockIdx.x * blockDim.x + threadIdx.x;
    const int v = (int)(tid >> 5);
    const int lane = threadIdx.x & 31;
    if (v >= N_VOCAB) return;
    ((float4*)(out + (size_t)v * DIM))[lane] =
        ((const float4*)(h + (size_t)idx[v] * DIM))[lane];
}

// ---------------------------------------------------------------------------
static void run_layer(const float* x_in, const int* edges, const float* W,
                      const float* a_src, const float* a_dst, const float* bias,
                      float* xw, float* h_out, float* s_src, float* s_dst,
                      unsigned* m_ord, float* ssum, hipStream_t stream) {
    const int ET = N_EDGES + N_NODES;
    gemm_xw_kernel<<<N_NODES / 16, 256, 0, stream>>>(x_in, W, xw);
    proj_kernel<<<(N_NODES * 32) / 256, 256, 0, stream>>>(xw, a_src, a_dst, s_src, s_dst);
    init_kernel<<<(N_NODES * DIM + 255) / 256, 256, 0, stream>>>(h_out, bias, m_ord, ssum);
    const int* esrc = edges;
    const int* edst = edges + N_EDGES;
    edge_max_kernel<<<(ET + 255) / 256, 256, 0, stream>>>(esrc, edst, s_src, s_dst, m_ord);
    edge_sum_kernel<<<(ET + 255) / 256, 256, 0, stream>>>(esrc, edst, s_src, s_dst, m_ord, ssum);
    edge_msg_kernel<<<(int)(((long)ET * 32 + 255) / 256), 256, 0, stream>>>(
        esrc, edst, s_src, s_dst, m_ord, ssum, xw, h_out);
}

extern "C" void kernel_launch(void* const* d_in, const int* in_sizes, int n_in,
                              void* d_out, int out_size, void* d_ws, size_t ws_size,
                              hipStream_t stream) {
    const float* embedding = (const float*)d_in[0];  // [N, 128]
    const float* W         = (const float*)d_in[1];  // [128, 128]
    const float* a_src     = (const float*)d_in[2];  // [128]
    const float* a_dst     = (const float*)d_in[3];  // [128]
    const float* bias      = (const float*)d_in[4];  // [128]
    const int*   edges1    = (const int*)d_in[5];    // [2, E]
    const int*   edges2    = (const int*)d_in[6];    // [2, E]
    const int*   idx_map   = (const int*)d_in[7];    // [V]
    float* out = (float*)d_out;                      // [V, 128]

    // workspace carve-up (~104 MB)
    float*    xw    = (float*)d_ws;                       // [N,128]
    float*    h     = xw + (size_t)N_NODES * DIM;         // [N,128]
    float*    s_src = h + (size_t)N_NODES * DIM;          // [N]
    float*    s_dst = s_src + N_NODES;                    // [N]
    unsigned* m_ord = (unsigned*)(s_dst + N_NODES);       // [N]
    float*    ssum  = (float*)(m_ord + N_NODES);          // [N]

    // layer 1: embedding -> h
    run_layer(embedding, edges1, W, a_src, a_dst, bias,
              xw, h, s_src, s_dst, m_ord, ssum, stream);
    // layer 2: h -> h  (h is consumed by the GEMM before init re-seeds it)
    run_layer(h, edges2, W, a_src, a_dst, bias,
              xw, h, s_src, s_dst, m_ord, ssum, stream);

    // out = h[idx_mapping]
    gather_kernel<<<(int)(((long)N_VOCAB * 32 + 255) / 256), 256, 0, stream>>>(
        idx_map, h, out);
}